// GCN4_63780264346288
// MI455X (gfx1250) — compile-verified
//
#include <hip/hip_runtime.h>
#include <hip/hip_bf16.h>

// ---------------------------------------------------------------------------
// GCNConv forward on MI455X (gfx1250):
//   xw = x @ W            (WMMA f32 16x16x4)
//   deg/dis               (atomic degree, rsqrt)
//   agg[dst] += xw[src] * dis[src]*dis[dst]   (wave-per-edge atomic scatter)
//   out = (agg + xw/deg)/deg + b ; log_softmax rows   (wave-per-node)
// ---------------------------------------------------------------------------

typedef __attribute__((ext_vector_type(2))) float v2f;
typedef __attribute__((ext_vector_type(8))) float v8f;

#define GCN_F 256
#define GCN_C 64

// ---- deg[i] = 1.0 (self-loop) ---------------------------------------------
__global__ void gcn_init_deg(float* __restrict__ deg, int N) {
    int i = blockIdx.x * blockDim.x + threadIdx.x;
    if (i < N) deg[i] = 1.0f;
}

// ---- deg[dst] += 1 per edge -----------------------------------------------
__global__ void gcn_deg_accum(const int* __restrict__ ei, float* __restrict__ deg, int E) {
    int e = blockIdx.x * blockDim.x + threadIdx.x;
    if (e < E) {
        int d = ei[E + e];              // edge_index[1] = dst
        atomicAdd(&deg[d], 1.0f);
    }
}

// ---- dis[i] = rsqrt(deg[i]) -----------------------------------------------
__global__ void gcn_dis(const float* __restrict__ deg, float* __restrict__ dis, int N) {
    int i = blockIdx.x * blockDim.x + threadIdx.x;
    if (i < N) dis[i] = rsqrtf(deg[i]);
}

// ---- xw = x @ W via V_WMMA_F32_16X16X4_F32 --------------------------------
// grid.x = N/16 (16-row block per block), 4 waves/block, wave w -> col tile w.
// A frag (16x4 f32, 2 VGPR): lanes 0-15 = rows, K = {0,1}; lanes 16-31 K = {2,3}.
// B frag (4x16 f32, 2 VGPR): lanes 0-15 = cols, K = {0,1}; lanes 16-31 K = {2,3}.
// C/D (16x16 f32, 8 VGPR): elem(M = r + 8*(lane>=16), N = lane&15) in VGPR r.
__global__ void gcn_gemm_wmma(const float* __restrict__ x, const float* __restrict__ W,
                              float* __restrict__ xw, int N) {
    const int lane  = threadIdx.x & 31;
    const int wave  = threadIdx.x >> 5;          // 0..3 -> 16-col tile
    const int hi    = lane >> 4;                 // 0 or 1
    const int l15   = lane & 15;
    const int kk    = hi << 1;                   // 0 or 2

    int row = blockIdx.x * 16 + l15;
    if (row >= N) row = N - 1;                   // clamp, keep EXEC all-ones
    const int col = wave * 16 + l15;

    const float* __restrict__ xrow = x + (size_t)row * GCN_F;
    v8f acc = {0.f, 0.f, 0.f, 0.f, 0.f, 0.f, 0.f, 0.f};

    #pragma unroll 4
    for (int k = 0; k < GCN_F; k += 4) {
        v2f a, bb;
        a.x  = xrow[k + kk];
        a.y  = xrow[k + kk + 1];
        bb.x = W[(k + kk)     * GCN_C + col];
        bb.y = W[(k + kk + 1) * GCN_C + col];
        acc = __builtin_amdgcn_wmma_f32_16x16x4_f32(
            /*neg_a=*/false, a, /*neg_b=*/false, bb,
            /*c_mod=*/(short)0, acc, /*reuse_a=*/false, /*reuse_b=*/false);
    }

    const int rowBase = blockIdx.x * 16 + (hi << 3);
    #pragma unroll
    for (int r = 0; r < 8; ++r) {
        int rr = rowBase + r;
        if (rr < N) xw[(size_t)rr * GCN_C + col] = acc[r];
    }
}

// ---- agg[dst] += xw[src] * dis[src]*dis[dst], one wave per edge -----------
__global__ void gcn_scatter(const int* __restrict__ ei, const float* __restrict__ xw,
                            const float* __restrict__ dis, float* __restrict__ agg, int E) {
    const int lane   = threadIdx.x & 31;
    const int wib    = threadIdx.x >> 5;
    const int wpb    = blockDim.x >> 5;
    const int nwaves = gridDim.x * wpb;
    for (int e = blockIdx.x * wpb + wib; e < E; e += nwaves) {
        const int s = ei[e];
        const int d = ei[E + e];
        const float coef = dis[s] * dis[d];
        const float* __restrict__ xs = xw + (size_t)s * GCN_C;
        float* __restrict__ ad = agg + (size_t)d * GCN_C;
        atomicAdd(&ad[lane],      xs[lane]      * coef);
        atomicAdd(&ad[lane + 32], xs[lane + 32] * coef);
    }
}

// ---- out = (agg + xw/deg)/deg + b ; log_softmax(row), one wave per node ---
__global__ void gcn_finalize(float* __restrict__ out, const float* __restrict__ xw,
                             const float* __restrict__ deg, const float* __restrict__ b, int N) {
    const int lane = threadIdx.x & 31;
    const int node = blockIdx.x * (blockDim.x >> 5) + (threadIdx.x >> 5);
    if (node >= N) return;

    const float inv  = 1.0f / deg[node];
    const size_t base = (size_t)node * GCN_C;

    float a0 = (out[base + lane]      + xw[base + lane]      * inv) * inv + b[lane];
    float a1 = (out[base + lane + 32] + xw[base + lane + 32] * inv) * inv + b[lane + 32];

    float m = fmaxf(a0, a1);
    #pragma unroll
    for (int off = 16; off > 0; off >>= 1) m = fmaxf(m, __shfl_xor(m, off, 32));

    float s = expf(a0 - m) + expf(a1 - m);
    #pragma unroll
    for (int off = 16; off > 0; off >>= 1) s += __shfl_xor(s, off, 32);

    const float lse = logf(s) + m;
    out[base + lane]      = a0 - lse;
    out[base + lane + 32] = a1 - lse;
}

extern "C" void kernel_launch(void* const* d_in, const int* in_sizes, int n_in,
                              void* d_out, int out_size, void* d_ws, size_t ws_size,
                              hipStream_t stream) {
    const float* x  = (const float*)d_in[0];   // [N, 256]
    const int*   ei = (const int*)d_in[1];     // [2, E]
    const float* W  = (const float*)d_in[2];   // [256, 64]
    const float* b  = (const float*)d_in[3];   // [64]

    const int N = in_sizes[0] / GCN_F;
    const int E = in_sizes[1] / 2;

    float* out = (float*)d_out;                // also used as agg accumulator
    float* deg = (float*)d_ws;                 // [N]
    float* dis = deg + N;                      // [N]
    float* xw  = dis + N;                      // [N, 64]

    // agg = 0 (d_out doubles as agg)
    hipMemsetAsync(d_out, 0, (size_t)N * GCN_C * sizeof(float), stream);

    gcn_init_deg<<<(N + 255) / 256, 256, 0, stream>>>(deg, N);
    gcn_deg_accum<<<(E + 255) / 256, 256, 0, stream>>>(ei, deg, E);
    gcn_dis<<<(N + 255) / 256, 256, 0, stream>>>(deg, dis, N);

    gcn_gemm_wmma<<<(N + 15) / 16, 128, 0, stream>>>(x, W, xw, N);

    gcn_scatter<<<4096, 256, 0, stream>>>(ei, xw, dis, out, E);

    gcn_finalize<<<(N + 7) / 8, 256, 0, stream>>>(out, xw, deg, b, N);
}